// CaNet_12970801234197
// MI455X (gfx1250) — compile-verified
//
#include <hip/hip_runtime.h>
#include <math.h>

typedef __attribute__((ext_vector_type(2))) float v2f;
typedef __attribute__((ext_vector_type(8))) float v8f;

#define N_NODES  50000
#define HID      128
#define D_IN     128
#define N_CLS    47
#define K_ENV    4
#define N_TILES16 (N_NODES / 16)          // 3125, exact
#define N_TILES32 ((N_NODES + 31) / 32)   // 1563, last tile is half

__device__ __forceinline__ void atomic_add_f32(float* p, float v) {
    __hip_atomic_fetch_add(p, v, __ATOMIC_RELAXED, __HIP_MEMORY_SCOPE_AGENT);
}

// ---------------------------------------------------------------- utilities
__global__ void fill_zero_kernel(float* __restrict__ p, long n) {
    long i = (long)blockIdx.x * blockDim.x + threadIdx.x;
    if (i < n) p[i] = 0.0f;
}

__global__ void degree_kernel(const int* __restrict__ ei, int E, float* __restrict__ deg) {
    int e = blockIdx.x * blockDim.x + threadIdx.x;
    if (e < E) {
        int col = ei[E + e];
        atomic_add_f32(&deg[col], 1.0f);
    }
}

__global__ void dinv_kernel(float* __restrict__ deg) {
    int n = blockIdx.x * blockDim.x + threadIdx.x;
    if (n < N_NODES) {
        float d = deg[n];
        deg[n] = (d > 0.0f) ? (1.0f / sqrtf(d)) : 0.0f;
    }
}

// ------------------------------------------------- fc0: h = relu(x@W + b)
// One wave per 32-node tile: 2 A fragments share every B fragment
// -> 16 v_wmma_f32_16x16x4_f32 per 16 B loads.
__global__ __launch_bounds__(128) void fc0_kernel(const float* __restrict__ x,
                                                  const float* __restrict__ w,
                                                  const float* __restrict__ b,
                                                  float* __restrict__ h) {
    int wave = threadIdx.x >> 5;
    int tile = blockIdx.x * 4 + wave;
    if (tile >= N_TILES32) return;
    int l  = threadIdx.x & 31;
    int m  = l & 15;
    int kb = (l >> 4) * 2;
    int node0 = tile * 32;
    int r1 = node0 + 16 + m;
    int r1c = (r1 < N_NODES) ? r1 : (N_NODES - 1);      // clamp tail loads
    const float* xrow0 = x + (size_t)(node0 + m) * D_IN;
    const float* xrow1 = x + (size_t)r1c * D_IN;

    v8f acc0[8], acc1[8];
#pragma unroll
    for (int o = 0; o < 8; ++o) { acc0[o] = (v8f){}; acc1[o] = (v8f){}; }

    for (int t = 0; t < D_IN / 4; ++t) {
        int d = t * 4 + kb;
        v2f a0, a1;
        a0.x = xrow0[d];  a0.y = xrow0[d + 1];
        a1.x = xrow1[d];  a1.y = xrow1[d + 1];
        const float* wr0 = w + (size_t)d * HID;
        const float* wr1 = wr0 + HID;
#pragma unroll
        for (int o = 0; o < 8; ++o) {
            v2f bb;
            bb.x = wr0[o * 16 + m];
            bb.y = wr1[o * 16 + m];
            acc0[o] = __builtin_amdgcn_wmma_f32_16x16x4_f32(false, a0, false, bb,
                                                            (short)0, acc0[o], false, false);
            acc1[o] = __builtin_amdgcn_wmma_f32_16x16x4_f32(false, a1, false, bb,
                                                            (short)0, acc1[o], false, false);
        }
    }
    int rowoff = (l >> 4) * 8;
#pragma unroll
    for (int o = 0; o < 8; ++o) {
        int f = o * 16 + m;
        float bias = b[f];
#pragma unroll
        for (int v = 0; v < 8; ++v) {
            int n0 = node0 + v + rowoff;
            h[(size_t)n0 * HID + f] = fmaxf(acc0[o][v] + bias, 0.0f);
            int n1 = node0 + 16 + v + rowoff;
            if (n1 < N_NODES)
                h[(size_t)n1 * HID + f] = fmaxf(acc1[o][v] + bias, 0.0f);
        }
    }
}

// -------------------------------------- env gate: e = softmax(h@ew + eb), K=4
__global__ void env_kernel(const float* __restrict__ h, const float* __restrict__ ew,
                           const float* __restrict__ eb, float* __restrict__ e) {
    int n = blockIdx.x * blockDim.x + threadIdx.x;
    if (n >= N_NODES) return;
    float s0 = eb[0], s1 = eb[1], s2 = eb[2], s3 = eb[3];
    const float* hr = h + (size_t)n * HID;
    for (int d = 0; d < HID; ++d) {
        float hv = hr[d];
        s0 += hv * ew[d * 4 + 0];
        s1 += hv * ew[d * 4 + 1];
        s2 += hv * ew[d * 4 + 2];
        s3 += hv * ew[d * 4 + 3];
    }
    float mx = fmaxf(fmaxf(s0, s1), fmaxf(s2, s3));
    s0 = expf(s0 - mx); s1 = expf(s1 - mx); s2 = expf(s2 - mx); s3 = expf(s3 - mx);
    float inv = 1.0f / (s0 + s1 + s2 + s3);
    e[(size_t)n * 4 + 0] = s0 * inv;
    e[(size_t)n * 4 + 1] = s1 * inv;
    e[(size_t)n * 4 + 2] = s2 * inv;
    e[(size_t)n * 4 + 3] = s3 * inv;
}

// ------------------- GCN scatter: agg[col] += dinv[row]*dinv[col] * h[row]
// One wave32 per edge; each lane owns 4 consecutive features (float4 gather).
__global__ __launch_bounds__(256) void scatter_kernel(const int* __restrict__ ei, int E,
                                                      const float* __restrict__ dinv,
                                                      const float* __restrict__ h,
                                                      float* __restrict__ agg) {
    int wave = threadIdx.x >> 5;
    int e = blockIdx.x * 8 + wave;
    if (e >= E) return;
    int l = threadIdx.x & 31;
    int row = ei[e];
    int col = ei[E + e];
    float val = dinv[row] * dinv[col];
    const float4* src = (const float4*)(h + (size_t)row * HID);
    float4 v = src[l];
    float* dst = agg + (size_t)col * HID + l * 4;
    atomic_add_f32(dst + 0, val * v.x);
    atomic_add_f32(dst + 1, val * v.y);
    atomic_add_f32(dst + 2, val * v.z);
    atomic_add_f32(dst + 3, val * v.w);
}

// --- mixture conv: h = relu( sum_{k,d} (e[n,k]*hi[n,d]) * W[k,d,:] + h[n] )
// hi = [agg | h] (2*HID); gate folded into A => single K=1024 WMMA GEMM.
// 32-node tile per wave (2 A frags x 8 output tiles = 16 wmma / K-step).
__global__ __launch_bounds__(128) void conv_kernel(const float* __restrict__ W,
                                                   const float* __restrict__ e,
                                                   const float* __restrict__ agg,
                                                   float* __restrict__ h) {
    int wave = threadIdx.x >> 5;
    int tile = blockIdx.x * 4 + wave;
    if (tile >= N_TILES32) return;
    int l  = threadIdx.x & 31;
    int m  = l & 15;
    int kb = (l >> 4) * 2;
    int node0 = tile * 32;
    int r0 = node0 + m;
    int r1 = node0 + 16 + m;
    int r1c = (r1 < N_NODES) ? r1 : (N_NODES - 1);      // clamp tail loads
    const float* aggrow0 = agg + (size_t)r0 * HID;
    const float* hrow0   = h   + (size_t)r0 * HID;
    const float* aggrow1 = agg + (size_t)r1c * HID;
    const float* hrow1   = h   + (size_t)r1c * HID;

    v8f acc0[8], acc1[8];
#pragma unroll
    for (int o = 0; o < 8; ++o) { acc0[o] = (v8f){}; acc1[o] = (v8f){}; }

    for (int kenv = 0; kenv < K_ENV; ++kenv) {
        float es0 = e[(size_t)r0 * 4 + kenv];
        float es1 = e[(size_t)r1c * 4 + kenv];
        const float* Wk = W + (size_t)kenv * (2 * HID) * HID;
        for (int t = 0; t < (2 * HID) / 4; ++t) {       // d = 0..255 step 4
            int d = t * 4 + kb;                          // d,d+1 never straddle HID
            const float* s0 = (d < HID) ? aggrow0 : (hrow0 - HID);
            const float* s1 = (d < HID) ? aggrow1 : (hrow1 - HID);
            v2f a0, a1;
            a0.x = es0 * s0[d];  a0.y = es0 * s0[d + 1];
            a1.x = es1 * s1[d];  a1.y = es1 * s1[d + 1];
            const float* wr0 = Wk + (size_t)d * HID;
            const float* wr1 = wr0 + HID;
#pragma unroll
            for (int o = 0; o < 8; ++o) {
                v2f bb;
                bb.x = wr0[o * 16 + m];
                bb.y = wr1[o * 16 + m];
                acc0[o] = __builtin_amdgcn_wmma_f32_16x16x4_f32(false, a0, false, bb,
                                                                (short)0, acc0[o], false, false);
                acc1[o] = __builtin_amdgcn_wmma_f32_16x16x4_f32(false, a1, false, bb,
                                                                (short)0, acc1[o], false, false);
            }
        }
    }
    // residual + relu, in-place (each wave owns its rows exclusively)
    int rowoff = (l >> 4) * 8;
#pragma unroll
    for (int o = 0; o < 8; ++o) {
        int f = o * 16 + m;
#pragma unroll
        for (int v = 0; v < 8; ++v) {
            int n0 = node0 + v + rowoff;
            size_t i0 = (size_t)n0 * HID + f;
            h[i0] = fmaxf(acc0[o][v] + h[i0], 0.0f);
            int n1 = node0 + 16 + v + rowoff;
            if (n1 < N_NODES) {
                size_t i1 = (size_t)n1 * HID + f;
                h[i1] = fmaxf(acc1[o][v] + h[i1], 0.0f);
            }
        }
    }
}

// ------------------------------------------------- fc1: out = h@W + b  (47 cols)
__global__ __launch_bounds__(128) void fc1_kernel(const float* __restrict__ h,
                                                  const float* __restrict__ w,
                                                  const float* __restrict__ b,
                                                  float* __restrict__ out) {
    int wave = threadIdx.x >> 5;
    int tile = blockIdx.x * 4 + wave;
    if (tile >= N_TILES16) return;
    int l  = threadIdx.x & 31;
    int m  = l & 15;
    int kb = (l >> 4) * 2;
    int node0 = tile * 16;
    const float* hrow = h + (size_t)(node0 + m) * HID;

    v8f acc[3];
#pragma unroll
    for (int o = 0; o < 3; ++o) acc[o] = (v8f){};

    for (int t = 0; t < HID / 4; ++t) {
        int d = t * 4 + kb;
        v2f a;
        a.x = hrow[d];
        a.y = hrow[d + 1];
        const float* wr0 = w + (size_t)d * N_CLS;
        const float* wr1 = wr0 + N_CLS;
#pragma unroll
        for (int o = 0; o < 3; ++o) {
            int f = o * 16 + m;
            v2f bb;
            bb.x = (f < N_CLS) ? wr0[f] : 0.0f;
            bb.y = (f < N_CLS) ? wr1[f] : 0.0f;
            acc[o] = __builtin_amdgcn_wmma_f32_16x16x4_f32(false, a, false, bb,
                                                           (short)0, acc[o], false, false);
        }
    }
    int rowoff = (l >> 4) * 8;
#pragma unroll
    for (int o = 0; o < 3; ++o) {
        int f = o * 16 + m;
        if (f < N_CLS) {
            float bias = b[f];
#pragma unroll
            for (int v = 0; v < 8; ++v) {
                int node = node0 + v + rowoff;
                out[(size_t)node * N_CLS + f] = acc[o][v] + bias;
            }
        }
    }
}

// ---------------------------------------------------------------- launcher
extern "C" void kernel_launch(void* const* d_in, const int* in_sizes, int n_in,
                              void* d_out, int out_size, void* d_ws, size_t ws_size,
                              hipStream_t stream) {
    const float* x      = (const float*)d_in[0];
    const int*   ei     = (const int*)  d_in[1];   // [2, E]
    const float* fc0_w  = (const float*)d_in[2];
    const float* fc0_b  = (const float*)d_in[3];
    const float* fc1_w  = (const float*)d_in[4];
    const float* fc1_b  = (const float*)d_in[5];
    const float* env_w  = (const float*)d_in[6];   // [L,128,4]
    const float* env_b  = (const float*)d_in[7];   // [L,4]
    const float* conv_w = (const float*)d_in[8];   // [L,4,256,128]
    float*       out    = (float*)d_out;

    const int E = in_sizes[1] / 2;

    // workspace layout (floats): h | agg | e | deg   (~52 MB)
    float* h   = (float*)d_ws;
    float* agg = h   + (size_t)N_NODES * HID;
    float* ev  = agg + (size_t)N_NODES * HID;
    float* deg = ev  + (size_t)N_NODES * K_ENV;

    const int blocks32 = (N_TILES32 + 3) / 4;   // 32-row tiles, 4 waves/block
    const int blocks16 = (N_TILES16 + 3) / 4;   // 16-row tiles, 4 waves/block

    // degrees -> dinv (in place)
    fill_zero_kernel<<<(N_NODES + 255) / 256, 256, 0, stream>>>(deg, N_NODES);
    degree_kernel<<<(E + 255) / 256, 256, 0, stream>>>(ei, E, deg);
    dinv_kernel<<<(N_NODES + 255) / 256, 256, 0, stream>>>(deg);

    // h = relu(x @ fc0_w + fc0_b)
    fc0_kernel<<<blocks32, 128, 0, stream>>>(x, fc0_w, fc0_b, h);

    for (int i = 0; i < 2; ++i) {
        env_kernel<<<(N_NODES + 255) / 256, 256, 0, stream>>>(
            h, env_w + (size_t)i * HID * K_ENV, env_b + (size_t)i * K_ENV, ev);

        fill_zero_kernel<<<((long)N_NODES * HID + 255) / 256, 256, 0, stream>>>(
            agg, (long)N_NODES * HID);
        scatter_kernel<<<(E + 7) / 8, 256, 0, stream>>>(ei, E, deg, h, agg);

        conv_kernel<<<blocks32, 128, 0, stream>>>(
            conv_w + (size_t)i * K_ENV * 2 * HID * HID, ev, agg, h);
    }

    // out = h @ fc1_w + fc1_b
    fc1_kernel<<<blocks16, 128, 0, stream>>>(h, fc1_w, fc1_b, out);
}